// YOLOv1Loss_42305427865908
// MI455X (gfx1250) — compile-verified
//
#include <hip/hip_runtime.h>
#include <math.h>

// ---------------- problem constants ----------------
#define S_DIM 7
#define B_BOX 2
#define C_CLS 20
#define CELL_PRED 30   // B*5+C floats per cell in predictions
#define CELL_TGT  25   // 4+1+C  floats per cell in targets
#define EPS_F 1e-6f
#define CPB   256      // cells per block == threads per block
#define THREADS 256

typedef __attribute__((ext_vector_type(4))) float v4f;
typedef __attribute__((ext_vector_type(4))) int   v4i;

#define AS_GLOBAL __attribute__((address_space(1)))
#define AS_LDS    __attribute__((address_space(3)))

// ---------------- CDNA5 async global->LDS path ----------------
#if __has_builtin(__builtin_amdgcn_global_load_async_to_lds_b128)
#define HAS_ASYNC_B128 1
#endif

__device__ __forceinline__ void copy_b128_to_lds(const float* g, float* l) {
#if defined(HAS_ASYNC_B128)
    // emits GLOBAL_LOAD_ASYNC_TO_LDS_B128 (tracked by ASYNCcnt)
    __builtin_amdgcn_global_load_async_to_lds_b128(
        (AS_GLOBAL v4i*)g, (AS_LDS v4i*)l, /*offset=*/0, /*cpol=*/0);
#else
    *(v4f*)l = *(const v4f*)g;   // fallback: through-VGPR copy
#endif
}

__device__ __forceinline__ void wait_async_copies() {
#if defined(HAS_ASYNC_B128)
#if __has_builtin(__builtin_amdgcn_s_wait_asynccnt)
    __builtin_amdgcn_s_wait_asynccnt(0);
#else
    asm volatile("s_wait_asynccnt 0" ::: "memory");
#endif
#endif
}

// ---------------- main per-cell kernel ----------------
__global__ void __launch_bounds__(THREADS)
yolo_main_kernel(const float* __restrict__ pred,
                 const float* __restrict__ tgt,
                 float* __restrict__ partials,
                 int nCells)
{
    __shared__ float lp[CPB * CELL_PRED];   // 30720 B
    __shared__ float lt[CPB * CELL_TGT];    // 25600 B
    __shared__ float lred[8 * 4];

    const int t = threadIdx.x;
    const int base = blockIdx.x * CPB;
    const int cellsHere = (nCells - base < CPB) ? (nCells - base) : CPB;

    const float* gp = pred + (long long)base * CELL_PRED;
    const float* gq = tgt  + (long long)base * CELL_TGT;

    // Stage predictions: (CPB*30)/4 = 1920 16B chunks
    const int nPredChunks = (cellsHere * CELL_PRED) >> 2;
#pragma unroll
    for (int i = 0; i < 8; ++i) {
        int idx = t + i * THREADS;
        if (idx < nPredChunks) copy_b128_to_lds(gp + idx * 4, &lp[idx * 4]);
    }
    // Stage targets: (CPB*25)/4 = 1600 16B chunks
    const int nTgtChunks = (cellsHere * CELL_TGT) >> 2;
#pragma unroll
    for (int i = 0; i < 7; ++i) {
        int idx = t + i * THREADS;
        if (idx < nTgtChunks) copy_b128_to_lds(gq + idx * 4, &lt[idx * 4]);
    }
    wait_async_copies();
    __syncthreads();

    float acc0 = 0.f, acc1 = 0.f, acc2 = 0.f, acc3 = 0.f;

    const int cid = base + t;
    if (cid < nCells) {
        const float* p = &lp[t * CELL_PRED];
        const float* q = &lt[t * CELL_TGT];

        const int rem = cid % (S_DIM * S_DIM);
        const float fi = (float)(rem / S_DIM);   // row -> y
        const float fj = (float)(rem % S_DIM);   // col -> x

        // ground-truth box -> absolute xyxy (exact reference op order)
        const float txv = q[0], tyv = q[1], twv = q[2], thv = q[3];
        const float objf = (q[4] > 0.0f) ? 1.0f : 0.0f;
        const float gcx = (fj + txv) / 7.0f;
        const float gcy = (fi + tyv) / 7.0f;
        const float gx1 = gcx - twv / 2.0f, gy1 = gcy - thv / 2.0f;
        const float gx2 = gcx + twv / 2.0f, gy2 = gcy + thv / 2.0f;
        const float garea = (gx2 - gx1) * (gy2 - gy1);

        float bx[B_BOX], by[B_BOX], bw[B_BOX], bh[B_BOX], cf[B_BOX], iou[B_BOX];
#pragma unroll
        for (int k = 0; k < B_BOX; ++k) {
            bx[k] = p[5 * k + 0]; by[k] = p[5 * k + 1];
            bw[k] = p[5 * k + 2]; bh[k] = p[5 * k + 3];
            cf[k] = p[5 * k + 4];
            const float pcx = (fj + bx[k]) / 7.0f;
            const float pcy = (fi + by[k]) / 7.0f;
            const float px1 = pcx - bw[k] / 2.0f, py1 = pcy - bh[k] / 2.0f;
            const float px2 = pcx + bw[k] / 2.0f, py2 = pcy + bh[k] / 2.0f;
            const float ix1 = fmaxf(px1, gx1), iy1 = fmaxf(py1, gy1);
            const float ix2 = fminf(px2, gx2), iy2 = fminf(py2, gy2);
            const float inter = fmaxf(ix2 - ix1, 0.0f) * fmaxf(iy2 - iy1, 0.0f);
            const float parea = (px2 - px1) * (py2 - py1);
            iou[k] = inter / (parea + garea - inter + EPS_F);
        }

        // jnp.argmax picks first max -> best=1 only when strictly greater
        const int best = (iou[1] > iou[0]) ? 1 : 0;
        const float rx = bx[best], ry = by[best], rw = bw[best], rh = bh[best];
        const float rconf = cf[best], riou = iou[best];

        // coord loss
        const float dx = rx - txv, dy = ry - tyv;
        const float sw = sqrtf(rw + EPS_F) - sqrtf(twv + EPS_F);
        const float sh = sqrtf(rh + EPS_F) - sqrtf(thv + EPS_F);
        acc0 = objf * (dx * dx + dy * dy + sw * sw + sh * sh);

        // obj confidence loss
        const float dc = rconf - riou;
        acc1 = objf * dc * dc;

        // no-obj confidence loss
        const float m0 = objf * ((best == 0) ? 1.0f : 0.0f);
        const float m1 = objf * ((best == 1) ? 1.0f : 0.0f);
        acc2 = (1.0f - m0) * cf[0] * cf[0] + (1.0f - m1) * cf[1] * cf[1];

        // class loss: -objf * sum(gt_cls * log_softmax(logits))
        const float* lg = p + B_BOX * 5;
        float mx = lg[0];
#pragma unroll
        for (int c = 1; c < C_CLS; ++c) mx = fmaxf(mx, lg[c]);
        float se = 0.0f;
#pragma unroll
        for (int c = 0; c < C_CLS; ++c) se += expf(lg[c] - mx);
        const float lse = mx + logf(se);
        float dot = 0.0f;
#pragma unroll
        for (int c = 0; c < C_CLS; ++c) dot += q[5 + c] * (lse - lg[c]);
        acc3 = objf * dot;
    }

    // wave32 butterfly reduction
#pragma unroll
    for (int off = 16; off > 0; off >>= 1) {
        acc0 += __shfl_xor(acc0, off, 32);
        acc1 += __shfl_xor(acc1, off, 32);
        acc2 += __shfl_xor(acc2, off, 32);
        acc3 += __shfl_xor(acc3, off, 32);
    }
    const int wave = t >> 5, lane = t & 31;
    if (lane == 0) {
        lred[wave * 4 + 0] = acc0;
        lred[wave * 4 + 1] = acc1;
        lred[wave * 4 + 2] = acc2;
        lred[wave * 4 + 3] = acc3;
    }
    __syncthreads();
    if (t == 0) {
        float s0 = 0.f, s1 = 0.f, s2 = 0.f, s3 = 0.f;
#pragma unroll
        for (int w = 0; w < 8; ++w) {
            s0 += lred[w * 4 + 0];
            s1 += lred[w * 4 + 1];
            s2 += lred[w * 4 + 2];
            s3 += lred[w * 4 + 3];
        }
        float* o = partials + (size_t)blockIdx.x * 4;
        o[0] = s0; o[1] = s1; o[2] = s2; o[3] = s3;
    }
}

// ---------------- deterministic final reduction ----------------
__global__ void __launch_bounds__(256)
yolo_finalize_kernel(const float* __restrict__ partials, int nBlocks,
                     float* __restrict__ out, float inv_n)
{
    __shared__ float l0[256], l1[256], l2[256], l3[256];
    const int t = threadIdx.x;
    float s0 = 0.f, s1 = 0.f, s2 = 0.f, s3 = 0.f;
    for (int g = t; g < nBlocks; g += 256) {
        s0 += partials[g * 4 + 0];
        s1 += partials[g * 4 + 1];
        s2 += partials[g * 4 + 2];
        s3 += partials[g * 4 + 3];
    }
    l0[t] = s0; l1[t] = s1; l2[t] = s2; l3[t] = s3;
    __syncthreads();
    for (int stride = 128; stride > 0; stride >>= 1) {
        if (t < stride) {
            l0[t] += l0[t + stride];
            l1[t] += l1[t + stride];
            l2[t] += l2[t + stride];
            l3[t] += l3[t + stride];
        }
        __syncthreads();
    }
    if (t == 0) {
        const float c0 = l0[0] * inv_n;   // loss_coord / n
        const float c1 = l1[0] * inv_n;   // loss_conf_obj / n
        const float c2 = l2[0] * inv_n;   // loss_conf_noobj / n
        const float c3 = l3[0] * inv_n;   // loss_class / n
        out[0] = c0;
        out[1] = c1;
        out[2] = c2;
        out[3] = c3;
        out[4] = 5.0f * c0 + c1 + 0.5f * c2 + c3;  // LAMBDA_COORD, LAMBDA_NOOBJ
    }
}

// ---------------- launcher ----------------
extern "C" void kernel_launch(void* const* d_in, const int* in_sizes, int n_in,
                              void* d_out, int out_size, void* d_ws, size_t ws_size,
                              hipStream_t stream) {
    const float* pred = (const float*)d_in[0];
    const float* tgt  = (const float*)d_in[1];
    float* out        = (float*)d_out;
    float* partials   = (float*)d_ws;

    const long long nPredElems = (long long)in_sizes[0];
    const int nCells  = (int)(nPredElems / CELL_PRED);             // 802816
    const int bs      = nCells / (S_DIM * S_DIM);                  // 16384
    const int nBlocks = (nCells + CPB - 1) / CPB;                  // 3136

    yolo_main_kernel<<<nBlocks, THREADS, 0, stream>>>(pred, tgt, partials, nCells);
    yolo_finalize_kernel<<<1, 256, 0, stream>>>(partials, nBlocks, out,
                                                1.0f / (float)bs);
}